// Self_Attention_64166811402453
// MI455X (gfx1250) — compile-verified
//
#include <hip/hip_runtime.h>

typedef float v2f __attribute__((ext_vector_type(2)));
typedef float v8f __attribute__((ext_vector_type(8)));

#define NNODES 8192
#define MEDGES 2048
#define KMEM   16
#define QKDIM  256
#define NHEAD  8
#define NEGF   (-9000000000000000.0f)
#define NBLKS  (NNODES / 16)          // 512 node blocks
#define ECAP   1024                   // max edges per node block (mean ~64)

__device__ __forceinline__ v8f wmma4(v2f a, v2f b, v8f c) {
  // D = A(16x4 f32) * B(4x16 f32) + C(16x16 f32)
  return __builtin_amdgcn_wmma_f32_16x16x4_f32(false, a, false, b, (short)0, c,
                                               false, false);
}

// ---------------------------------------------------------------------------
// Kernel 1: single coalesced pass over dense adj (N x M).
//  - collect the 16 member node ids of each hyperedge (order fixed by sort)
//  - mark touched edges in a per-16-node-block bitmap (order-independent)
// ---------------------------------------------------------------------------
__global__ __launch_bounds__(256) void build_members(const float* __restrict__ adj,
                                                     int* __restrict__ cnt,
                                                     int* __restrict__ idx,
                                                     unsigned int* __restrict__ bitmap) {
  size_t g = (size_t)blockIdx.x * 256 + threadIdx.x;   // g in [0, N*M)
  float a = adj[g];
  if (a > 0.0f) {
    int m = (int)(g & (MEDGES - 1));
    int n = (int)(g >> 11);                            // M = 2048 = 2^11
    int pos = atomicAdd(&cnt[m], 1);
    if (pos < KMEM) idx[m * KMEM + pos] = n;
    int blk = n >> 4;
    atomicOr(&bitmap[blk * (MEDGES / 32) + (m >> 5)], 1u << (m & 31));
  }
}

// ---------------------------------------------------------------------------
// Kernel 2: hyperedge weights. One block (128 thr) per hyperedge.
// d2(i,j) = G_ii + G_jj - 2 G_ij with G = F F^T computed by fp32 WMMA.
// sigma = lower median (rank 59 of 120), w = sum(exp(-d2/sigma^2))/240.
// ---------------------------------------------------------------------------
__global__ __launch_bounds__(128) void edge_weights(const float* __restrict__ x1,
                                                    const int* __restrict__ idx,
                                                    float* __restrict__ w) {
  __shared__ float F[KMEM * QKDIM];   // 16 KB gathered features
  __shared__ float G[KMEM * KMEM];    // Gram
  __shared__ int   snode[KMEM];
  __shared__ float d2s[128];
  __shared__ float red[128];
  __shared__ float sigma_s;

  const int m = blockIdx.x;
  const int tid = threadIdx.x;

  if (tid == 0) {                     // sort member ids -> determinism
    int t[KMEM];
    for (int i = 0; i < KMEM; ++i) t[i] = idx[m * KMEM + i];
    for (int i = 1; i < KMEM; ++i) {
      int key = t[i], j = i - 1;
      while (j >= 0 && t[j] > key) { t[j + 1] = t[j]; --j; }
      t[j + 1] = key;
    }
    for (int i = 0; i < KMEM; ++i) snode[i] = t[i];
  }
  __syncthreads();

  for (int q4 = tid; q4 < KMEM * (QKDIM / 4); q4 += 128) {
    int row = q4 >> 6;
    int c4  = (q4 & 63) << 2;
    float4 s = *(const float4*)(x1 + (size_t)snode[row] * QKDIM + c4);
    *(float4*)(&F[row * QKDIM + c4]) = s;
  }
  __syncthreads();

  if (tid < 32) {                     // wave 0: G = F F^T, 64 fp32 WMMA steps
    const int lane = tid, half = lane >> 4, lr = lane & 15;
    v8f c = {0.f, 0.f, 0.f, 0.f, 0.f, 0.f, 0.f, 0.f};
    const float* frow = &F[lr * QKDIM + half * 2];
    for (int k0 = 0; k0 < QKDIM; k0 += 4) {
      v2f ab = *(const v2f*)(frow + k0);
      c = wmma4(ab, ab, c);
    }
    for (int r = 0; r < 8; ++r) G[(r + half * 8) * KMEM + lr] = c[r];
  }
  __syncthreads();

  if (tid < 120) {                    // 120 unique pairs
    int t = tid, i = 0;
    while (t >= 15 - i) { t -= 15 - i; ++i; }
    int j = i + 1 + t;
    d2s[tid] = G[i * KMEM + i] + G[j * KMEM + j] - 2.0f * G[i * KMEM + j];
  }
  __syncthreads();

  if (tid < 120) {                    // rank selection -> sorted[59]
    float mine = d2s[tid];
    int rank = 0;
    for (int j = 0; j < 120; ++j) {
      float o = d2s[j];
      rank += (o < mine) || (o == mine && j < tid);
    }
    if (rank == 59) sigma_s = mine;
  }
  __syncthreads();

  float e = 0.0f;
  if (tid < 120) {
    float sg = sigma_s;
    e = expf(-d2s[tid] / (sg * sg));
  }
  red[tid] = e;
  __syncthreads();
  for (int s = 64; s > 0; s >>= 1) {
    if (tid < s) red[tid] += red[tid + s];
    __syncthreads();
  }
  if (tid == 0) w[m] = red[0] * (1.0f / 240.0f);
}

// ---------------------------------------------------------------------------
// Kernel 2b: fallback vector fall[c] = sum_m w[m]*v[m][c]   (c in [0,256))
// Used for nodes with zero hyperedges (reference: uniform softmax over M).
// ---------------------------------------------------------------------------
__global__ __launch_bounds__(256) void fall_vec(const float* __restrict__ v,
                                                const float* __restrict__ w,
                                                float* __restrict__ fall) {
  int c = threadIdx.x;
  float acc = 0.0f;
  for (int m = 0; m < MEDGES; ++m) acc += w[m] * v[(size_t)m * QKDIM + c];
  fall[c] = acc;
}

// ---------------------------------------------------------------------------
// Kernel 2c: compact each node-block bitmap into an ascending edge list.
// Deterministic (independent of kernel-1 atomic ordering).
// ---------------------------------------------------------------------------
__global__ __launch_bounds__(64) void compact_lists(const unsigned int* __restrict__ bitmap,
                                                    int* __restrict__ elist,
                                                    int* __restrict__ ecount) {
  __shared__ int cnts[64];
  __shared__ int offs[64];
  const int blk = blockIdx.x, t = threadIdx.x;
  unsigned int bits = bitmap[blk * (MEDGES / 32) + t];
  cnts[t] = __popc(bits);
  __syncthreads();
  if (t == 0) {
    int acc = 0;
    for (int i = 0; i < 64; ++i) { offs[i] = acc; acc += cnts[i]; }
    ecount[blk] = acc < ECAP ? acc : ECAP;
  }
  __syncthreads();
  int off = offs[t];
  for (int b = 0; b < 32; ++b)
    if ((bits >> b) & 1u) {
      if (off < ECAP) elist[blk * ECAP + off] = t * 32 + b;
      ++off;
    }
}

// ---------------------------------------------------------------------------
// Kernel 3: sparse masked flash attention over each block's edge list.
// One block = 16 node rows; 8 waves = 8 heads sharing gathered k/v/adj tiles.
// fp32 WMMA for q*k^T and P*v. Exact vs reference: masked entries underflow
// to 0 whenever the row has a member; memberless rows take the fallback path.
// ---------------------------------------------------------------------------
__global__ __launch_bounds__(256) void flash_attn(const float* __restrict__ x1,
                                                  const float* __restrict__ x2,
                                                  const float* __restrict__ v,
                                                  const float* __restrict__ adj,
                                                  const float* __restrict__ w,
                                                  const float* __restrict__ fall,
                                                  const int* __restrict__ elist,
                                                  const int* __restrict__ ecount,
                                                  float* __restrict__ out) {
  __shared__ float kt[KMEM * QKDIM];          // 16 KB: gathered k rows
  __shared__ float vt[KMEM * QKDIM];          // 16 KB: gathered w-scaled v rows
  __shared__ float adjt[KMEM * KMEM];         // mask tile
  __shared__ float pscr[NHEAD][KMEM * KMEM];  // P relayout scratch
  __shared__ int   me[KMEM];                  // edge ids of current tile
  __shared__ int   s_cnt;

  const int nb   = blockIdx.x;
  const int tid  = threadIdx.x;
  const int wave = tid >> 5;                  // head id
  const int lane = tid & 31;
  const int half = lane >> 4;
  const int lr   = lane & 15;
  const int hbase = wave * 32;

  if (tid == 0) s_cnt = ecount[nb];

  // q tile as WMMA A-operand
  v2f aq[8];
  {
    const float scale = 0.17677669529663687f; // 32^-0.5
    const float* qp = x1 + (size_t)(nb * 16 + lr) * QKDIM + hbase + half * 2;
    for (int s = 0; s < 8; ++s) {
      v2f t = *(const v2f*)(qp + s * 4);
      aq[s].x = t.x * scale;
      aq[s].y = t.y * scale;
    }
  }

  v8f out0 = {0.f, 0.f, 0.f, 0.f, 0.f, 0.f, 0.f, 0.f};
  v8f out1 = {0.f, 0.f, 0.f, 0.f, 0.f, 0.f, 0.f, 0.f};
  float rmax[8], rsum[8];
  for (int r = 0; r < 8; ++r) { rmax[r] = -__builtin_inff(); rsum[r] = 0.0f; }

  __syncthreads();
  const int ntiles = (s_cnt + 15) >> 4;

  for (int t = 0; t < ntiles; ++t) {
    __syncthreads();  // previous tile's LDS reads complete
    if (tid < 16) {
      int j = t * 16 + tid;
      me[tid] = (j < s_cnt) ? elist[nb * ECAP + j] : -1;
    }
    __syncthreads();

    // gather k rows and w-scaled v rows (zeros for padded columns)
    for (int q4 = tid; q4 < KMEM * (QKDIM / 4); q4 += 256) {
      int row = q4 >> 6;
      int c4  = (q4 & 63) << 2;
      int mrow = me[row];
      float4 kk = {0.f, 0.f, 0.f, 0.f}, vv = {0.f, 0.f, 0.f, 0.f};
      if (mrow >= 0) {
        kk = *(const float4*)(x2 + (size_t)mrow * QKDIM + c4);
        float ww = w[mrow];
        vv = *(const float4*)(v + (size_t)mrow * QKDIM + c4);
        vv.x *= ww; vv.y *= ww; vv.z *= ww; vv.w *= ww;
      }
      *(float4*)(&kt[row * QKDIM + c4]) = kk;
      *(float4*)(&vt[row * QKDIM + c4]) = vv;
    }
    {
      int r = tid >> 4, c = tid & 15;
      int mc = me[c];
      adjt[tid] = (mc >= 0) ? adj[(size_t)(nb * 16 + r) * MEDGES + mc] : 0.0f;
    }
    __syncthreads();

    // S = q * k^T  (16 nodes x 16 listed edges), 8 WMMA steps
    v8f S = {0.f, 0.f, 0.f, 0.f, 0.f, 0.f, 0.f, 0.f};
    for (int s = 0; s < 8; ++s) {
      v2f b = *(const v2f*)(&kt[lr * QKDIM + hbase + s * 4 + half * 2]);
      S = wmma4(aq[s], b, S);
    }

    // mask + online softmax (row r + half*8 lives across a 16-lane group)
    for (int r = 0; r < 8; ++r) {
      float sv = (adjt[(r + half * 8) * 16 + lr] > 0.0f) ? S[r] : NEGF;
      float mx = sv;
      mx = fmaxf(mx, __shfl_xor(mx, 8));
      mx = fmaxf(mx, __shfl_xor(mx, 4));
      mx = fmaxf(mx, __shfl_xor(mx, 2));
      mx = fmaxf(mx, __shfl_xor(mx, 1));
      float nmax = fmaxf(rmax[r], mx);
      float sc   = expf(rmax[r] - nmax);
      rmax[r] = nmax;
      float p = expf(sv - nmax);
      float ps = p;
      ps += __shfl_xor(ps, 8);
      ps += __shfl_xor(ps, 4);
      ps += __shfl_xor(ps, 2);
      ps += __shfl_xor(ps, 1);
      rsum[r] = rsum[r] * sc + ps;
      out0[r] *= sc;
      out1[r] *= sc;
      pscr[wave][(r + half * 8) * 16 + lr] = p;   // C-layout -> LDS
    }
    __syncthreads();  // P visible across lanes (block stays convergent)

    // out += P (16x16) * vv (16x32): 4 WMMA steps x 2 dv tiles
    for (int kc = 0; kc < 4; ++kc) {
      v2f ap = *(const v2f*)(&pscr[wave][lr * 16 + kc * 4 + half * 2]);
      int mi = (kc * 4 + half * 2) * QKDIM + hbase + lr;
      v2f b0, b1;
      b0.x = vt[mi];      b0.y = vt[mi + QKDIM];
      b1.x = vt[mi + 16]; b1.y = vt[mi + QKDIM + 16];
      out0 = wmma4(ap, b0, out0);
      out1 = wmma4(ap, b1, out1);
    }
  }

  // epilogue: normalize, or fallback (uniform over all M) for memberless rows
  for (int r = 0; r < 8; ++r) {
    int row = nb * 16 + r + half * 8;
    bool dead = rmax[r] < -1.0e14f;   // member scores are O(10)
    float inv = dead ? 0.0f : 1.0f / rsum[r];
    float o0 = dead ? fall[hbase + lr]      * (1.0f / MEDGES) : out0[r] * inv;
    float o1 = dead ? fall[hbase + 16 + lr] * (1.0f / MEDGES) : out1[r] * inv;
    out[(size_t)row * QKDIM + hbase + lr]      = o0;
    out[(size_t)row * QKDIM + hbase + 16 + lr] = o1;
  }
}

// ---------------------------------------------------------------------------
extern "C" void kernel_launch(void* const* d_in, const int* in_sizes, int n_in,
                              void* d_out, int out_size, void* d_ws, size_t ws_size,
                              hipStream_t stream) {
  const float* x1  = (const float*)d_in[0];
  const float* x2  = (const float*)d_in[1];
  const float* v   = (const float*)d_in[2];
  const float* adj = (const float*)d_in[3];
  // d_in[4] edge_attr: recomputed, unused
  float* out = (float*)d_out;

  // workspace layout
  char* ws = (char*)d_ws;
  size_t off = 0;
  int*          cnt    = (int*)(ws + off);          off += MEDGES * 4;          // 8 KB
  int*          idx    = (int*)(ws + off);          off += MEDGES * KMEM * 4;   // 128 KB
  float*        w      = (float*)(ws + off);        off += MEDGES * 4;          // 8 KB
  float*        fall   = (float*)(ws + off);        off += QKDIM * 4;           // 1 KB
  size_t bm_off = off;
  unsigned int* bitmap = (unsigned int*)(ws + off); off += NBLKS * (MEDGES / 32) * 4; // 128 KB
  size_t bm_sz = off - bm_off;
  int*          ecount = (int*)(ws + off);          off += NBLKS * 4;           // 2 KB
  int*          elist  = (int*)(ws + off);          off += NBLKS * ECAP * 4;    // 2 MB

  hipMemsetAsync(ws, 0, (MEDGES + MEDGES * KMEM) * 4, stream);  // cnt + idx
  hipMemsetAsync(ws + bm_off, 0, bm_sz, stream);                // bitmap

  hipLaunchKernelGGL(build_members, dim3((NNODES * MEDGES) / 256), dim3(256),
                     0, stream, adj, cnt, idx, bitmap);
  hipLaunchKernelGGL(edge_weights, dim3(MEDGES), dim3(128), 0, stream,
                     x1, idx, w);
  hipLaunchKernelGGL(fall_vec, dim3(1), dim3(256), 0, stream, v, w, fall);
  hipLaunchKernelGGL(compact_lists, dim3(NBLKS), dim3(64), 0, stream,
                     bitmap, elist, ecount);
  hipLaunchKernelGGL(flash_attn, dim3(NBLKS), dim3(256), 0, stream,
                     x1, x2, v, adj, w, fall, elist, ecount, out);
}